// custom_loss_58402965291305
// MI455X (gfx1250) — compile-verified
//
#include <hip/hip_runtime.h>
#include <stdint.h>

// ---------------- problem constants (B,H,W,C = 32,512,512,3) ----------------
#define ROWF     1536          // W*C floats per image row
#define IMGF     786432        // H*ROWF floats per image
#define OUTR0    5             // first output row (crop)
#define NOUTR    502           // output rows per image (5..506)
#define COLLO    15            // first valid flat column index (w=5,c=0)
#define COLHI    1520          // last  valid flat column index (w=506,c=2)
#define RCHUNK   32            // output rows per workgroup
#define CHUNKS   16            // ceil(502/32)
#define NTHREADS 384           // 12 wave32s; 384 * float4 = one 6144B row
#define K0F      0.30780133f
#define K1F      0.38439734f
#define INVCNT   (1.0f/756012.0f)                       // 502*502*3
#define C1F      ((float)(1.0/(255.0*255.0*255.0*255.0))) // S^4  (quadratic term)
#define C2F      ((float)(-2.0/(255.0*255.0)))            // -2*S^2 (linear term)

typedef float v4f __attribute__((ext_vector_type(4)));
typedef int   v4i __attribute__((ext_vector_type(4)));

// ---------------- CDNA5 async global->LDS helpers ----------------
typedef __attribute__((address_space(1))) v4i* gv4p;   // global src (builtin is non-const)
typedef __attribute__((address_space(3))) v4i* lv4p;   // LDS dst

__device__ __forceinline__ void async_cp16(const float* g, float* l) {
#if __has_builtin(__builtin_amdgcn_global_load_async_to_lds_b128)
    __builtin_amdgcn_global_load_async_to_lds_b128(
        (gv4p)(uintptr_t)g, (lv4p)(uint32_t)(uintptr_t)l, 0, 0);
#else
    asm volatile("global_load_async_to_lds_b128 %0, %1, off"
                 :: "v"((uint32_t)(uintptr_t)l), "v"(g) : "memory");
#endif
}

__device__ __forceinline__ void wait_async_le2() {
#if __has_builtin(__builtin_amdgcn_s_wait_asynccnt)
    __builtin_amdgcn_s_wait_asynccnt(2);
#else
    asm volatile("s_wait_asynccnt 0x2" ::: "memory");
#endif
}
__device__ __forceinline__ void wait_async_le0() {
#if __has_builtin(__builtin_amdgcn_s_wait_asynccnt)
    __builtin_amdgcn_s_wait_asynccnt(0);
#else
    asm volatile("s_wait_asynccnt 0x0" ::: "memory");
#endif
}

// 3-tap vertical filter, K0 == K2: add + mul + fma
__device__ __forceinline__ v4f tap3(v4f m, v4f c, v4f p) {
    return K0F * (m + p) + K1F * c;
}

// ---------------- stage 1: per-workgroup streaming filter + partial sum ----
__global__ __launch_bounds__(NTHREADS)
void filt_loss_kernel(const float* __restrict__ X, const float* __restrict__ Y,
                      float* __restrict__ partial) {
    __shared__ float sx[5][ROWF];
    __shared__ float sy[5][ROWF];
    __shared__ float wsum[NTHREADS / 32];

    const int tid   = threadIdx.x;
    const int blk   = blockIdx.x;
    const int b     = blk >> 4;          // image index (CHUNKS == 16)
    const int chunk = blk & (CHUNKS - 1);

    const int out_row0  = OUTR0 + chunk * RCHUNK;
    const int rem       = (OUTR0 + NOUTR) - out_row0;       // rows left
    const int nrows_out = rem < RCHUNK ? rem : RCHUNK;
    const int in_row0   = out_row0 - 1;
    const int rows_in   = nrows_out + 2;

    const float* gx = X + (size_t)b * IMGF;
    const float* gy = Y + (size_t)b * IMGF;
    const int fo = tid * 4;  // this thread's float4 segment within a row

    // crop mask: flat column j valid iff 15 <= j <= 1520 (w in 5..506)
    v4f mask;
    mask.x = (fo + 0 >= COLLO && fo + 0 <= COLHI) ? 1.0f : 0.0f;
    mask.y = (fo + 1 >= COLLO && fo + 1 <= COLHI) ? 1.0f : 0.0f;
    mask.z = (fo + 2 >= COLLO && fo + 2 <= COLHI) ? 1.0f : 0.0f;
    mask.w = (fo + 3 >= COLLO && fo + 3 <= COLHI) ? 1.0f : 0.0f;

    // prologue: stage input rows 0,1,2 of this chunk
    for (int i = 0; i < 3; ++i) {
        const size_t ro = (size_t)(in_row0 + i) * ROWF + fo;
        async_cp16(gx + ro, &sx[i][fo]);
        async_cp16(gy + ro, &sy[i][fo]);
    }

    v4f accq = {0.0f, 0.0f, 0.0f, 0.0f};   // sum of t1*t2   (quadratic term)
    v4f accl = {0.0f, 0.0f, 0.0f, 0.0f};   // sum of t3      (linear term)
    int s0 = 0, s1 = 1, s2 = 2, sn = 3;    // circular slots: rows o,o+1,o+2,next

    for (int o = 0; o < nrows_out; ++o) {
        const int nr = o + 3;
        if (nr < rows_in) {               // prefetch row o+3 (overwrites row o-2 slot)
            const size_t ro = (size_t)(in_row0 + nr) * ROWF + fo;
            async_cp16(gx + ro, &sx[sn][fo]);
            async_cp16(gy + ro, &sy[sn][fo]);
            wait_async_le2();             // row o+2 pair now complete
        } else {
            wait_async_le0();
        }
        __syncthreads();                  // rows o..o+2 visible to all waves

        // one aligned float4 per row per tensor (ds_load_b128); raw (unscaled) data
        const v4f am = *(const v4f*)(sx[s0] + fo);
        const v4f ac = *(const v4f*)(sx[s1] + fo);
        const v4f ap = *(const v4f*)(sx[s2] + fo);
        const v4f bm = *(const v4f*)(sy[s0] + fo);
        const v4f bc = *(const v4f*)(sy[s1] + fo);
        const v4f bp = *(const v4f*)(sy[s2] + fo);

        const v4f P   = tap3(am, ac, ap);             // F(x)
        const v4f Q   = tap3(bm, bc, bp);             // F(y)
        const v4f Pxx = tap3(am * am, ac * ac, ap * ap); // F(x*x)
        const v4f Qyy = tap3(bm * bm, bc * bc, bp * bp); // F(y*y)
        const v4f Rxy = tap3(am * bm, ac * bc, ap * bp); // F(x*y)

        const v4f t1 = Pxx - P * P;                   // sigma1_sq / S^2
        const v4f t2 = Qyy - Q * Q;                   // sigma2_sq / S^2
        const v4f t3 = Rxy - P * Q;                   // sigma12  / S^2

        const v4f u = t1 * t2;
        accq += mask * u;                             // fma(mask, u, accq)
        accl += mask * t3;                            // fma(mask, t3, accl)

        s0 = s1; s1 = s2; s2 = sn; sn = (sn == 4) ? 0 : sn + 1;
    }

    // loss = S^4 * sum(t1*t2) - 2*S^2 * sum(t3); constants applied once per thread
    float acc = C1F * (accq.x + accq.y + accq.z + accq.w)
              + C2F * (accl.x + accl.y + accl.z + accl.w);

    // wave32 shuffle reduction
    for (int m = 16; m > 0; m >>= 1) acc += __shfl_xor(acc, m, 32);
    if ((tid & 31) == 0) wsum[tid >> 5] = acc;
    __syncthreads();
    if (tid == 0) {
        float s = 0.0f;
        for (int i = 0; i < NTHREADS / 32; ++i) s += wsum[i];
        partial[blk] = s;
    }
}

// ---------------- stage 2: deterministic fixed-order final reduction -------
__global__ __launch_bounds__(512)
void reduce_kernel(const float* __restrict__ partial, float* __restrict__ out, int n) {
    __shared__ float s[512];
    const int t = threadIdx.x;
    s[t] = (t < n) ? partial[t] : 0.0f;
    __syncthreads();
    for (int k = 256; k > 0; k >>= 1) {
        if (t < k) s[t] += s[t + k];
        __syncthreads();
    }
    if (t == 0) out[0] = s[0] * INVCNT;
}

// ---------------- entry point ----------------
extern "C" void kernel_launch(void* const* d_in, const int* in_sizes, int n_in,
                              void* d_out, int out_size, void* d_ws, size_t ws_size,
                              hipStream_t stream) {
    const float* X = (const float*)d_in[0];
    const float* Y = (const float*)d_in[1];
    float* out     = (float*)d_out;
    float* partial = (float*)d_ws;           // B*CHUNKS floats of scratch

    const int B = in_sizes[0] / IMGF;        // 32 for the reference shapes
    const int nblocks = B * CHUNKS;          // 512

    filt_loss_kernel<<<dim3(nblocks), dim3(NTHREADS), 0, stream>>>(X, Y, partial);
    reduce_kernel<<<dim3(1), dim3(512), 0, stream>>>(partial, out, nblocks);
}